// TextureLoss_39152921870850
// MI455X (gfx1250) — compile-verified
//
#include <hip/hip_runtime.h>
#include <math.h>

// CDNA5 wave32 WMMA-based reduction + LBP texture loss.
// Shapes hard-coded to the reference: (16,3,512,512) f32, H=W=512 (power of 2
// -> pure shift/mask index math, no integer division).

typedef float v2f __attribute__((ext_vector_type(2)));
typedef float v8f __attribute__((ext_vector_type(8)));

#define NPLANES   48          // 16 * 3
#define TASKS     1572864     // 48 * 512 * 64
#define NBLOCKS   6144        // TASKS / 256

// ---------------------------------------------------------------------------
// Wave32 reduction of one float per lane via V_WMMA_F32_16X16X4_F32.
// A (16x4): lane L<16 holds A[L][0]=partial, A[L][1]=0; lane L>=16 holds
// A[L-16][2]=partial, A[L-16][3]=0.  B = all ones.  Then
//   D[m][n] = partial[m] + partial[m+16]  for every n.
// D layout: lane L, VGPR j -> row M = j + (L<16 ? 0 : 8).  So summing the 8 D
// registers gives lanes 0..15:  sum of rows 0..7, lanes 16..31: rows 8..15.
// lane0 + lane16 therefore carry the full 32-lane sum.  Both are stored to
// smem[wave*2 + {0,1}] so one thread can finish the block reduction.
// ---------------------------------------------------------------------------
__device__ __forceinline__ void block_reduce_store(float partial, float* smem16) {
    const unsigned lane = threadIdx.x & 31u;
    const unsigned wave = threadIdx.x >> 5;
#if defined(__gfx1250__) && __has_builtin(__builtin_amdgcn_wmma_f32_16x16x4_f32)
    v2f a; a[0] = partial; a[1] = 0.0f;
    v2f b; b[0] = 1.0f;    b[1] = 1.0f;
    v8f c = {};
    v8f d = __builtin_amdgcn_wmma_f32_16x16x4_f32(
        /*neg_a=*/false, a, /*neg_b=*/false, b,
        /*c_mod=*/(short)0, c, /*reuse_a=*/false, /*reuse_b=*/false);
    float t = ((d[0] + d[1]) + (d[2] + d[3])) + ((d[4] + d[5]) + (d[6] + d[7]));
    if (lane == 0u)  smem16[wave * 2u + 0u] = t;
    if (lane == 16u) smem16[wave * 2u + 1u] = t;
#else
    float t = partial;
    #pragma unroll
    for (int off = 16; off; off >>= 1) t += __shfl_xor(t, off, 32);
    if (lane == 0u) { smem16[wave * 2u + 0u] = t; smem16[wave * 2u + 1u] = 0.0f; }
#endif
}

// Load one padded row window: a[j] = row[x0 - 1 + j], j = 0..9, with
// reflect-101 at the image edge (col -1 -> col 1, col 512 -> col 510; those
// become pure register copies, no extra loads).
__device__ __forceinline__ void load_row(const float* __restrict__ row,
                                         unsigned x0, float a[10]) {
    const float4 v0 = *reinterpret_cast<const float4*>(row + x0);
    const float4 v1 = *reinterpret_cast<const float4*>(row + x0 + 4);
    a[1] = v0.x; a[2] = v0.y; a[3] = v0.z; a[4] = v0.w;
    a[5] = v1.x; a[6] = v1.y; a[7] = v1.z; a[8] = v1.w;
    a[0] = (x0 == 0u)   ? a[2] : row[x0 - 1];
    a[9] = (x0 == 504u) ? a[7] : row[x0 + 8];
}

// LBP code for output pixel i (center = m[i+1]); offset order matches the
// reference _OFFSETS -> bit weights 1,2,4,...,128.
__device__ __forceinline__ int lbp_code(const float t[10], const float m[10],
                                        const float b[10], int i) {
    const float c = m[i + 1];
    int code = 0;
    code |= (t[i]     >= c) ? 1   : 0;   // (-1,-1)
    code |= (t[i + 1] >= c) ? 2   : 0;   // (-1, 0)
    code |= (t[i + 2] >= c) ? 4   : 0;   // (-1, 1)
    code |= (m[i + 2] >= c) ? 8   : 0;   // ( 0, 1)
    code |= (b[i + 2] >= c) ? 16  : 0;   // ( 1, 1)
    code |= (b[i + 1] >= c) ? 32  : 0;   // ( 1, 0)
    code |= (b[i]     >= c) ? 64  : 0;   // ( 1,-1)
    code |= (m[i]     >= c) ? 128 : 0;   // ( 0,-1)
    return code;
}

// One task = 8 consecutive pixels of one row of one plane, both images.
__global__ __launch_bounds__(256)
void lbp_diff_kernel(const float* __restrict__ R, const float* __restrict__ F,
                     float* __restrict__ part, int atomicMode) {
    __shared__ float smem[16];

    const unsigned tid = blockIdx.x * 256u + threadIdx.x;
    const unsigned xg  = tid & 63u;            // x-group (8 px each)
    const unsigned y   = (tid >> 6) & 511u;    // row
    const unsigned p   = tid >> 15;            // plane (b*3 + c), 0..47
    const unsigned x0  = xg << 3;
    const unsigned base = p << 18;             // plane * 512*512
    const unsigned ym  = (y == 0u)   ? 1u   : y - 1u;   // reflect-101
    const unsigned yp  = (y == 511u) ? 510u : y + 1u;

    float tr[10], mr[10], br[10], tf[10], mf[10], bf[10];
    load_row(R + base + (ym << 9), x0, tr);
    load_row(R + base + (y  << 9), x0, mr);
    load_row(R + base + (yp << 9), x0, br);
    load_row(F + base + (ym << 9), x0, tf);
    load_row(F + base + (y  << 9), x0, mf);
    load_row(F + base + (yp << 9), x0, bf);

    unsigned acc = 0;   // exact: sum of |code_r - code_f| <= 8*255
    #pragma unroll
    for (int i = 0; i < 8; ++i) {
        const int cr = lbp_code(tr, mr, br, i);
        const int cf = lbp_code(tf, mf, bf, i);
        const int d  = cr - cf;
        acc += (unsigned)(d < 0 ? -d : d);
    }

    block_reduce_store((float)acc, smem);   // WMMA wave reduce (exact, < 2^24)
    __syncthreads();
    if (threadIdx.x == 0) {
        float s = 0.0f;
        #pragma unroll
        for (int i = 0; i < 16; ++i) s += smem[i];
        if (atomicMode) atomicAdd(part, s);
        else            part[blockIdx.x] = s;
    }
}

__global__ __launch_bounds__(256)
void reduce_final_kernel(const float* __restrict__ part, int n,
                         float* __restrict__ out) {
    __shared__ float smem[16];
    float s = 0.0f;
    for (int i = threadIdx.x; i < n; i += 256) s += part[i];  // uniform trip count
    block_reduce_store(s, smem);
    __syncthreads();
    if (threadIdx.x == 0) {
        float t = 0.0f;
        #pragma unroll
        for (int i = 0; i < 16; ++i) t += smem[i];
        out[0] = sqrtf(t * (1.0f / 255.0f));
    }
}

__global__ void zero_kernel(float* p) { p[0] = 0.0f; }

__global__ void sqrt_kernel(const float* __restrict__ acc, float* __restrict__ out) {
    out[0] = sqrtf(acc[0] * (1.0f / 255.0f));
}

extern "C" void kernel_launch(void* const* d_in, const int* in_sizes, int n_in,
                              void* d_out, int out_size, void* d_ws, size_t ws_size,
                              hipStream_t stream) {
    const float* R  = (const float*)d_in[0];   // real_image (16,3,512,512) f32
    const float* F  = (const float*)d_in[1];   // fake_image
    float*       out = (float*)d_out;          // scalar f32
    float*       ws  = (float*)d_ws;

    if (ws_size >= (size_t)NBLOCKS * sizeof(float)) {
        // Two-stage: per-block partials (no hot atomics), then 1-block reduce.
        lbp_diff_kernel<<<NBLOCKS, 256, 0, stream>>>(R, F, ws, 0);
        reduce_final_kernel<<<1, 256, 0, stream>>>(ws, NBLOCKS, out);
    } else {
        // Fallback: single f32 accumulator in ws[0], one atomic per block.
        zero_kernel<<<1, 1, 0, stream>>>(ws);
        lbp_diff_kernel<<<NBLOCKS, 256, 0, stream>>>(R, F, ws, 1);
        sqrt_kernel<<<1, 1, 0, stream>>>(ws, out);
    }
}